// GAT_78323023610401
// MI455X (gfx1250) — compile-verified
//
#include <hip/hip_runtime.h>
#include <hip/hip_bf16.h>
#include <math.h>

// ---------------------------------------------------------------------------
// GATv2 (2 layers) for MI455X / gfx1250.
//  - GEMMs via V_WMMA_F32_16X16X4_F32 (exact fp32 matrix core path)
//  - edge phase: gather/score/segment-softmax/scatter, sized to live in 192MB L2
// ---------------------------------------------------------------------------

typedef float v2f __attribute__((ext_vector_type(2)));
typedef float v8f __attribute__((ext_vector_type(8)));

#define N_NODES 50000
#define N_EDGES 500000
#define E_TOT   (N_EDGES + N_NODES)   // with self-loops
#define IN_CH   128
#define HID     32
#define HEADS   4
#define OUT_CH  64

// ---------------- order-preserving float <-> u32 for atomic max -------------
__device__ __forceinline__ unsigned f2ord(float f) {
  unsigned u = __float_as_uint(f);
  return (u & 0x80000000u) ? ~u : (u | 0x80000000u);
}
__device__ __forceinline__ float ord2f(unsigned u) {
  unsigned v = (u & 0x80000000u) ? (u & 0x7FFFFFFFu) : ~u;
  return __uint_as_float(v);
}
#define ORD_NEG_INF 0x007FFFFFu   // f2ord(-inf)

// ---------------- generic u32 fill ------------------------------------------
__global__ void fill_u32(unsigned* __restrict__ p, unsigned v, int n) {
  int i = blockIdx.x * blockDim.x + threadIdx.x;
  if (i < n) p[i] = v;
}

// ---------------- fused dual GEMM: Xl = A*Wl, Xr = A*Wr  (K = 128) ----------
// One wave per 16x16 output tile; NC/16 waves per block cover all column
// tiles of one 16-row tile. A operand is shared between the two WMMA chains.
template<int NC>
__global__ __launch_bounds__(32 * (NC / 16))
void dual_gemm_k128(const float* __restrict__ A,     // [nrows,128]
                    const float* __restrict__ Wl,    // [128,NC]
                    const float* __restrict__ Wr,    // [128,NC]
                    float* __restrict__ Xl,
                    float* __restrict__ Xr,
                    int nrows) {
  constexpr int K = 128;
  const int lane = threadIdx.x & 31;
  const int wave = threadIdx.x >> 5;
  const int row0 = blockIdx.x * 16;
  const int col0 = wave * 16;
  const int m  = lane & 15;     // row within A tile / col within B tile
  const int hi = lane >> 4;     // K half-select (0 -> K0/K1, 1 -> K2/K3)
  if (row0 >= nrows) return;

  const float* arow = A + (size_t)(row0 + m) * K;
  const int col = col0 + m;

  v8f accL = {};
  v8f accR = {};
#pragma unroll
  for (int kb = 0; kb < K / 4; ++kb) {
    const int k0 = kb * 4 + hi * 2;
    v2f a  = { arow[k0], arow[k0 + 1] };
    v2f bl = { Wl[(size_t)k0 * NC + col], Wl[(size_t)(k0 + 1) * NC + col] };
    v2f br = { Wr[(size_t)k0 * NC + col], Wr[(size_t)(k0 + 1) * NC + col] };
    accL = __builtin_amdgcn_wmma_f32_16x16x4_f32(false, a, false, bl,
                                                 (short)0, accL, false, false);
    accR = __builtin_amdgcn_wmma_f32_16x16x4_f32(false, a, false, br,
                                                 (short)0, accR, false, false);
  }
  // D layout: VGPR r holds rows r (lanes 0-15) and r+8 (lanes 16-31)
#pragma unroll
  for (int r = 0; r < 8; ++r) {
    const int row = row0 + r + hi * 8;
    Xl[(size_t)row * NC + col] = accL[r];
    Xr[(size_t)row * NC + col] = accR[r];
  }
}

// ---------------- per-edge attention score + segment max --------------------
// One wave per edge. Lane l covers VW consecutive channels of the H*CH vector.
template<int H, int CH>
__global__ __launch_bounds__(256)
void edge_score(const float* __restrict__ Xl, const float* __restrict__ Xr,
                const float* __restrict__ att, const int* __restrict__ ei,
                float* __restrict__ score, unsigned* __restrict__ smaxb) {
  constexpr int F  = H * CH;
  constexpr int VW = F / 32;
  constexpr int G  = 32 / H;      // lanes per head
  const int e = blockIdx.x * 8 + (threadIdx.x >> 5);
  if (e >= E_TOT) return;
  const int lane = threadIdx.x & 31;
  const int s = (e < N_EDGES) ? ei[e]           : (e - N_EDGES);
  const int d = (e < N_EDGES) ? ei[N_EDGES + e] : (e - N_EDGES);

  const float* xs = Xl + (size_t)s * F + lane * VW;
  const float* xd = Xr + (size_t)d * F + lane * VW;
  float acc = 0.f;
#pragma unroll
  for (int i = 0; i < VW; ++i) {
    float mm = xs[i] + xd[i];
    mm = (mm > 0.f) ? mm : 0.2f * mm;   // leaky_relu(0.2)
    acc += mm * att[lane * VW + i];
  }
#pragma unroll
  for (int off = 1; off < G; off <<= 1) acc += __shfl_xor(acc, off, 32);
  if ((lane & (G - 1)) == 0) {
    const int h = lane / G;
    score[(size_t)e * H + h] = acc;
    atomicMax(&smaxb[(size_t)d * H + h], f2ord(acc));
  }
}

// ---------------- exp(score - max), accumulate softmax denominators ---------
template<int H>
__global__ void edge_denom(const float* __restrict__ score,
                           const unsigned* __restrict__ smaxb,
                           const int* __restrict__ ei,
                           float* __restrict__ denom) {
  const int idx = blockIdx.x * blockDim.x + threadIdx.x;
  if (idx >= E_TOT * H) return;
  const int e = idx / H;
  const int h = idx - e * H;
  const int d = (e < N_EDGES) ? ei[N_EDGES + e] : (e - N_EDGES);
  const float p = expf(score[idx] - ord2f(smaxb[(size_t)d * H + h]));
  atomicAdd(&denom[(size_t)d * H + h], p);
}

// ---------------- out[dst] += alpha * Xl[src] -------------------------------
template<int H, int CH>
__global__ __launch_bounds__(256)
void edge_aggregate(const float* __restrict__ Xl,
                    const float* __restrict__ score,
                    const unsigned* __restrict__ smaxb,
                    const float* __restrict__ denom,
                    const int* __restrict__ ei,
                    float* __restrict__ out) {
  constexpr int F  = H * CH;
  constexpr int VW = F / 32;
  constexpr int G  = 32 / H;
  const int e = blockIdx.x * 8 + (threadIdx.x >> 5);
  if (e >= E_TOT) return;
  const int lane = threadIdx.x & 31;
  const int s = (e < N_EDGES) ? ei[e]           : (e - N_EDGES);
  const int d = (e < N_EDGES) ? ei[N_EDGES + e] : (e - N_EDGES);
  const int h = lane / G;

  const float sm = ord2f(smaxb[(size_t)d * H + h]);
  const float p  = expf(score[(size_t)e * H + h] - sm);
  const float alpha = p / (denom[(size_t)d * H + h] + 1e-16f);

  const float* xs = Xl + (size_t)s * F + lane * VW;
  float*       o  = out + (size_t)d * F + lane * VW;
#pragma unroll
  for (int i = 0; i < VW; ++i) atomicAdd(&o[i], alpha * xs[i]);
}

// ---------------- bias + activation -----------------------------------------
__global__ void bias_elu(float* __restrict__ v, const float* __restrict__ b,
                         int cmask, int n) {
  const int i = blockIdx.x * blockDim.x + threadIdx.x;
  if (i >= n) return;
  const float t = v[i] + b[i & cmask];
  v[i] = (t > 0.f) ? t : (expf(t) - 1.f);   // elu(alpha=1)
}
__global__ void bias_add(float* __restrict__ v, const float* __restrict__ b,
                         int cmask, int n) {
  const int i = blockIdx.x * blockDim.x + threadIdx.x;
  if (i >= n) return;
  v[i] = v[i] + b[i & cmask];
}

// ---------------------------------------------------------------------------
extern "C" void kernel_launch(void* const* d_in, const int* in_sizes, int n_in,
                              void* d_out, int out_size, void* d_ws, size_t ws_size,
                              hipStream_t stream) {
  (void)in_sizes; (void)n_in; (void)out_size; (void)ws_size;
  const float* x    = (const float*)d_in[0];
  const int*   ei   = (const int*)  d_in[1];
  const float* W1l  = (const float*)d_in[2];
  const float* W1r  = (const float*)d_in[3];
  const float* att1 = (const float*)d_in[4];
  const float* b1   = (const float*)d_in[5];
  const float* W2l  = (const float*)d_in[6];
  const float* W2r  = (const float*)d_in[7];
  const float* att2 = (const float*)d_in[8];
  const float* b2   = (const float*)d_in[9];
  float* out = (float*)d_out;

  // workspace layout (floats); layer-2 reuses layer-1 regions
  float*    xl1   = (float*)d_ws;                       // N*128
  float*    xr1   = xl1 + (size_t)N_NODES * 128;        // N*128
  float*    h     = xr1 + (size_t)N_NODES * 128;        // N*128 (agg -> elu)
  float*    score = h   + (size_t)N_NODES * 128;        // E_TOT*HEADS
  unsigned* smaxb = (unsigned*)(score + (size_t)E_TOT * HEADS);  // N*HEADS
  float*    denom = (float*)(smaxb + (size_t)N_NODES * HEADS);   // N*HEADS

  const int EB = (E_TOT + 7) / 8;      // edge-wave blocks (8 waves / block)

  // ========================= layer 1: 128 -> 4x32 ===========================
  dual_gemm_k128<128><<<N_NODES / 16, 256, 0, stream>>>(x, W1l, W1r, xl1, xr1, N_NODES);
  fill_u32<<<(N_NODES * 128 + 255) / 256, 256, 0, stream>>>((unsigned*)h, 0u, N_NODES * 128);
  fill_u32<<<(N_NODES * HEADS + 255) / 256, 256, 0, stream>>>(smaxb, ORD_NEG_INF, N_NODES * HEADS);
  fill_u32<<<(N_NODES * HEADS + 255) / 256, 256, 0, stream>>>((unsigned*)denom, 0u, N_NODES * HEADS);

  edge_score<HEADS, HID><<<EB, 256, 0, stream>>>(xl1, xr1, att1, ei, score, smaxb);
  edge_denom<HEADS><<<(E_TOT * HEADS + 255) / 256, 256, 0, stream>>>(score, smaxb, ei, denom);
  edge_aggregate<HEADS, HID><<<EB, 256, 0, stream>>>(xl1, score, smaxb, denom, ei, h);
  bias_elu<<<(N_NODES * 128 + 255) / 256, 256, 0, stream>>>(h, b1, 127, N_NODES * 128);

  // ========================= layer 2: 128 -> 64 =============================
  dual_gemm_k128<64><<<N_NODES / 16, 128, 0, stream>>>(h, W2l, W2r, xl1, xr1, N_NODES);
  fill_u32<<<(N_NODES * OUT_CH + 255) / 256, 256, 0, stream>>>((unsigned*)out, 0u, N_NODES * OUT_CH);
  fill_u32<<<(N_NODES + 255) / 256, 256, 0, stream>>>(smaxb, ORD_NEG_INF, N_NODES);
  fill_u32<<<(N_NODES + 255) / 256, 256, 0, stream>>>((unsigned*)denom, 0u, N_NODES);

  edge_score<1, OUT_CH><<<EB, 256, 0, stream>>>(xl1, xr1, att2, ei, score, smaxb);
  edge_denom<1><<<(E_TOT + 255) / 256, 256, 0, stream>>>(score, smaxb, ei, denom);
  edge_aggregate<1, OUT_CH><<<EB, 256, 0, stream>>>(xl1, score, smaxb, denom, ei, out);
  bias_add<<<(N_NODES * OUT_CH + 255) / 256, 256, 0, stream>>>(out, b2, 63, N_NODES * OUT_CH);
}